// Model_16389595201849
// MI455X (gfx1250) — compile-verified
//
#include <hip/hip_runtime.h>

typedef __attribute__((ext_vector_type(2))) float v2f;
typedef __attribute__((ext_vector_type(8))) float v8f;

#define N_NODES 50000
#define E_MSG_N 400000
#define E_TGT_N 50000
#define DIM 128
#define HDIM 128

// ---------------------------------------------------------------- zero ws ---
__global__ void zero_kernel(float4* __restrict__ p, int n4) {
    int i = blockIdx.x * blockDim.x + threadIdx.x;
    int stride = gridDim.x * blockDim.x;
    float4 z = make_float4(0.f, 0.f, 0.f, 0.f);
    for (; i < n4; i += stride) p[i] = z;
}

// ------------------------------------------------------- edge scatter-add ---
// One wave (32 lanes) per edge: lane loads float4 of the 128-float source row,
// scales by edge weight, atomically accumulates into acc[dst]. Lane 0 bumps
// the per-destination edge count (reference mean uses raw edge counts).
__global__ void __launch_bounds__(256)
scatter_kernel(const float* __restrict__ x_src,
               const int*   __restrict__ msg,   // [2, E_MSG_N] row0=src row1=dst
               const int*   __restrict__ tgt,   // [2, E_TGT_N]
               const float* __restrict__ w,     // [E_TGT_N]
               float* __restrict__ acc,         // [N_NODES, DIM]
               float* __restrict__ cnt)         // [N_NODES]
{
    int tid  = blockIdx.x * blockDim.x + threadIdx.x;
    int e    = tid >> 5;
    int lane = tid & 31;
    if (e >= E_MSG_N + E_TGT_N) return;

    int src, dst; float wt;
    if (e < E_MSG_N) {
        src = msg[e];
        dst = msg[E_MSG_N + e];
        wt  = 1.0f;
    } else {
        int et = e - E_MSG_N;
        src = tgt[et];
        dst = tgt[E_TGT_N + et];
        wt  = w[et];
    }

    const float4* xr = (const float4*)(x_src + (size_t)src * DIM);
    float4 v = xr[lane];                         // coalesced 512B row read
    float* ar = acc + (size_t)dst * DIM + lane * 4;
    atomicAdd(ar + 0, v.x * wt);
    atomicAdd(ar + 1, v.y * wt);
    atomicAdd(ar + 2, v.z * wt);
    atomicAdd(ar + 3, v.w * wt);
    if (lane == 0) atomicAdd(cnt + dst, 1.0f);
}

// --------------------------------------------------------- WMMA f32 GEMM ---
// out[m][n] = (acc[m]/max(cnt[m],1)) . Wrel[n] + x_dst[m] . Wroot[n] + b[n]
// One wave computes a 16x16 output tile via V_WMMA_F32_16X16X4_F32, K=128.
// A layout (16x4 f32): lanes 0-15 -> rows, vgpr0/1 = K,K+1 (lanes 16-31: K+2,K+3)
// B layout mirrors with lanes = output columns; B[k][n] = W[n][k] so each lane
// streams a contiguous f32 pair from row n of W. C/D: vgpr v -> row v+8*(lane/16).
__global__ void __launch_bounds__(256)
gemm_kernel(const float* __restrict__ acc,    // [N_NODES, DIM] raw sums
            const float* __restrict__ cnt,    // [N_NODES]
            const float* __restrict__ x_dst,  // [N_NODES, DIM]
            const float* __restrict__ Wrel,   // [HDIM, DIM]
            const float* __restrict__ Wroot,  // [HDIM, DIM]
            const float* __restrict__ bias,   // [HDIM]
            float* __restrict__ out)          // [N_NODES, HDIM]
{
    int wid  = (blockIdx.x * blockDim.x + threadIdx.x) >> 5; // 0..24999, no tail
    int lane = threadIdx.x & 31;
    int mtile = wid >> 3;     // 0..3124
    int ntile = wid & 7;      // 0..7
    int r  = lane & 15;
    int kh = lane >> 4;       // which K half-pair this lane supplies
    int m0 = mtile * 16;
    int n0 = ntile * 16;

    const float* arow  = acc   + (size_t)(m0 + r) * DIM + 2 * kh;
    const float* xrow  = x_dst + (size_t)(m0 + r) * DIM + 2 * kh;
    const float* wrel  = Wrel  + (size_t)(n0 + r) * DIM + 2 * kh;
    const float* wroot = Wroot + (size_t)(n0 + r) * DIM + 2 * kh;
    float inv = 1.0f / fmaxf(cnt[m0 + r], 1.0f);

    v8f c = {0.f, 0.f, 0.f, 0.f, 0.f, 0.f, 0.f, 0.f};
#pragma unroll
    for (int kk = 0; kk < DIM; kk += 4) {
        v2f a0 = *(const v2f*)(arow + kk);
        a0.x *= inv; a0.y *= inv;                 // fold mean into A
        v2f b0 = *(const v2f*)(wrel + kk);
        c = __builtin_amdgcn_wmma_f32_16x16x4_f32(false, a0, false, b0,
                                                  (short)0, c, false, false);
        v2f a1 = *(const v2f*)(xrow + kk);
        v2f b1 = *(const v2f*)(wroot + kk);
        c = __builtin_amdgcn_wmma_f32_16x16x4_f32(false, a1, false, b1,
                                                  (short)0, c, false, false);
    }

    float bv = bias[n0 + r];
#pragma unroll
    for (int v = 0; v < 8; ++v) {
        out[(size_t)(m0 + v + 8 * kh) * HDIM + n0 + r] = c[v] + bv;
    }
}

// ------------------------------------------------------------------ host ---
extern "C" void kernel_launch(void* const* d_in, const int* in_sizes, int n_in,
                              void* d_out, int out_size, void* d_ws, size_t ws_size,
                              hipStream_t stream) {
    const float* x_a      = (const float*)d_in[0];
    const float* x_b      = (const float*)d_in[1];
    const int*   msg_ab   = (const int*)  d_in[2];
    const int*   msg_ba   = (const int*)  d_in[3];
    const int*   tgt_ab   = (const int*)  d_in[4];
    const int*   tgt_ba   = (const int*)  d_in[5];
    const float* w_ab     = (const float*)d_in[6];
    const float* w_ba     = (const float*)d_in[7];
    const float* Wrel_ab  = (const float*)d_in[8];
    const float* Wroot_ab = (const float*)d_in[9];
    const float* b_ab     = (const float*)d_in[10];
    const float* Wrel_ba  = (const float*)d_in[11];
    const float* Wroot_ba = (const float*)d_in[12];
    const float* b_ba     = (const float*)d_in[13];
    float* out = (float*)d_out;

    float* ws    = (float*)d_ws;
    float* acc_a = ws;                                   // [N, D]
    float* acc_b = ws + (size_t)N_NODES * DIM;           // [N, D]
    float* cnt_a = ws + 2 * (size_t)N_NODES * DIM;       // [N]
    float* cnt_b = cnt_a + N_NODES;                      // [N]

    // Phase 0: zero accumulators + counts (12.9M floats, /4 exactly).
    size_t ztotal = 2 * (size_t)N_NODES * DIM + 2 * (size_t)N_NODES;
    int n4 = (int)(ztotal / 4);
    zero_kernel<<<2048, 256, 0, stream>>>((float4*)ws, n4);

    // Phase 1: edge scatter (wave per edge, exact grid: 450000*32/256 = 56250).
    int etot    = E_MSG_N + E_TGT_N;
    int sblocks = (etot * 32) / 256;
    scatter_kernel<<<sblocks, 256, 0, stream>>>(x_a, msg_ab, tgt_ab, w_ab,
                                                acc_b, cnt_b);  // a -> b
    scatter_kernel<<<sblocks, 256, 0, stream>>>(x_b, msg_ba, tgt_ba, w_ba,
                                                acc_a, cnt_a);  // b -> a

    // Phase 2: WMMA GEMM epilogue. 25000 waves per direction, 3125 blocks.
    int gblocks = ((N_NODES / 16) * (HDIM / 16) * 32) / 256;
    // out[0] = out_a (dst = a, weights *_ba)
    gemm_kernel<<<gblocks, 256, 0, stream>>>(acc_a, cnt_a, x_a,
                                             Wrel_ba, Wroot_ba, b_ba, out);
    // out[1] = out_b (dst = b, weights *_ab)
    gemm_kernel<<<gblocks, 256, 0, stream>>>(acc_b, cnt_b, x_b,
                                             Wrel_ab, Wroot_ab, b_ab,
                                             out + (size_t)N_NODES * HDIM);
}